// WARPLoss_29532195127619
// MI455X (gfx1250) — compile-verified
//
#include <hip/hip_runtime.h>
#include <hip/hip_bf16.h>
#include <stdint.h>

// WARP loss for MI455X (gfx1250, wave32).
//  K1: per-row stable compaction of negative scores + n_neg.
//  KW: margin_term via V_WMMA_F32_16X16X4_F32 (diag of input . sign^T).
//  K2: one wave32 per positive (i,j); 32 coalesced trials per step, ballot
//      early-exit; negative scores staged in LDS via async global->LDS.
//  K3: deterministic tree reduction of per-row losses to the scalar output.

#define B_   128
#define LBL_ 1024
#define T_   255

typedef int   v4i_ __attribute__((vector_size(16)));
typedef v4i_  __attribute__((address_space(1)))* as1_v4i_p;
typedef v4i_  __attribute__((address_space(3)))* as3_v4i_p;
typedef int   __attribute__((address_space(1)))* as1_i_p;
typedef int   __attribute__((address_space(3)))* as3_i_p;
typedef float v2f_ __attribute__((ext_vector_type(2)));
typedef float v8f_ __attribute__((ext_vector_type(8)));

// ---------------------------------------------------------------------------
// Kernel 1: per-row preprocessing (counts + stable compaction).
// ---------------------------------------------------------------------------
__global__ __launch_bounds__(256) void warp_prep(
    const float* __restrict__ inp, const int* __restrict__ tgt,
    float* __restrict__ neg_scores, int* __restrict__ n_neg) {
  const int i   = blockIdx.x;
  const int tid = threadIdx.x;              // 256 threads, 4 labels each
  __shared__ int s_cnt[256];

  float vals[4];
  int   tg[4];
  int   cnt = 0;
  const int base = i * LBL_ + tid * 4;
#pragma unroll
  for (int e = 0; e < 4; ++e) {
    vals[e] = inp[base + e];
    tg[e]   = tgt[base + e];
    if (tg[e] == 0) cnt++;
  }
  s_cnt[tid] = cnt;
  __syncthreads();

  // Hillis-Steele inclusive scan over per-thread negative counts.
  for (int off = 1; off < 256; off <<= 1) {
    int v = (tid >= off) ? s_cnt[tid - off] : 0;
    __syncthreads();
    s_cnt[tid] += v;
    __syncthreads();
  }
  const int excl = s_cnt[tid] - cnt;

  // Stable compaction of negative label scores.
  int pos = excl;
#pragma unroll
  for (int e = 0; e < 4; ++e)
    if (tg[e] == 0) neg_scores[(size_t)i * LBL_ + (pos++)] = vals[e];

  if (tid == 0) {
    const int nn = s_cnt[255];
    n_neg[i]     = nn;
    // Degenerate all-positive row: argsort is identity, sample index 0.
    if (nn == 0) neg_scores[(size_t)i * LBL_] = inp[(size_t)i * LBL_];
  }
}

// ---------------------------------------------------------------------------
// Kernel W: margin_term[i] = LBL + sum_k input[i,k]*sign[i,k] via WMMA.
// One wave32 per 16-row group. A = 16x4 f32 input tile (lane=M, VGPR+half=K),
// B = 4x16 f32 sign tile (lane=N, VGPR+half=K; mirrored layout), C/D = 16x16
// f32 accumulator over 256 K-chunks. The needed dots are the diagonal:
// D[m][m] lives in (lane m, VGPR m) for m<8 and (lane 16+m, VGPR m-8) for m>=8.
// ---------------------------------------------------------------------------
__global__ __launch_bounds__(32) void warp_margin(
    const float* __restrict__ inp, const int* __restrict__ tgt,
    float* __restrict__ margin_term) {
  const int lane = threadIdx.x;             // 0..31, one wave
  const int base = blockIdx.x * 16;         // 16 rows per wave, 8 blocks

#if defined(__gfx1250__) && __has_builtin(__builtin_amdgcn_wmma_f32_16x16x4_f32)
  const int r    = base + (lane & 15);      // row for both A (M) and B (N)
  const int khalf = (lane >> 4) << 1;       // lanes 0-15: K={0,1}; 16-31: K={2,3}
  const size_t rb = (size_t)r * LBL_;

  v8f_ c = {};
  for (int k0 = 0; k0 < LBL_; k0 += 4) {
    const int k = k0 + khalf;
    v2f_ a, b;
    a.x = inp[rb + k];
    a.y = inp[rb + k + 1];
    b.x = (tgt[rb + k]     == 0) ? 1.0f : -1.0f;
    b.y = (tgt[rb + k + 1] == 0) ? 1.0f : -1.0f;
    // D = A x B + C   (v_wmma_f32_16x16x4_f32)
    c = __builtin_amdgcn_wmma_f32_16x16x4_f32(
        /*neg_a=*/false, a, /*neg_b=*/false, b,
        /*c_mod=*/(short)0, c, /*reuse_a=*/false, /*reuse_b=*/false);
  }

  float carr[8];
#pragma unroll
  for (int e = 0; e < 8; ++e) carr[e] = c[e];

  if (lane < 8) {
    margin_term[base + lane] = (float)LBL_ + carr[lane];        // m = lane
  } else if (lane >= 24) {
    margin_term[base + (lane - 16)] = (float)LBL_ + carr[lane - 24]; // m = lane-16
  }
#else
  // Scalar fallback: lanes 0-15 each own one row.
  if (lane < 16) {
    const size_t rb = (size_t)(base + lane) * LBL_;
    float s = 0.0f;
    for (int k = 0; k < LBL_; ++k)
      s += (tgt[rb + k] == 0) ? inp[rb + k] : -inp[rb + k];
    margin_term[base + lane] = (float)LBL_ + s;
  }
#endif
}

// ---------------------------------------------------------------------------
// Kernel 2: trial scan. One block (8 wave32s) per row; ballot early-exit.
// ---------------------------------------------------------------------------
__global__ __launch_bounds__(256) void warp_main(
    const int* __restrict__ tgt, const float* __restrict__ inp,
    const float* __restrict__ u, const float* __restrict__ neg_scores,
    const float* __restrict__ margin_term, const int* __restrict__ n_neg,
    float* __restrict__ row_loss) {
  const int i    = blockIdx.x;
  const int tid  = threadIdx.x;
  const int lane = tid & 31;
  const int wv   = tid >> 5;                // 0..7

  __shared__ __align__(16) float s_neg[LBL_];   // compacted negative scores
  __shared__ float s_x[LBL_];                   // input row
  __shared__ int   s_t[LBL_];                   // target row
  __shared__ float s_rw[256];                   // rank-weight table
  __shared__ float s_part[8];

  const size_t rowbase = (size_t)i * LBL_;

  for (int j = tid; j < LBL_; j += 256) {
    s_x[j] = inp[rowbase + j];
    s_t[j] = tgt[rowbase + j];
  }

  // Stage compacted negative scores into LDS via CDNA5 async global->LDS
  // (ASYNCcnt-tracked, bypasses VGPRs); plain fallback otherwise.
#if defined(__gfx1250__) && __has_builtin(__builtin_amdgcn_global_load_async_to_lds_b128)
  {
    const int j0 = tid * 4;  // 1024 floats == 256 lanes * 16B, exactly once
    __builtin_amdgcn_global_load_async_to_lds_b128(
        (as1_v4i_p)(neg_scores + rowbase + j0),
        (as3_v4i_p)(&s_neg[j0]), 0, 0);
#if __has_builtin(__builtin_amdgcn_s_wait_asynccnt)
    __builtin_amdgcn_s_wait_asynccnt(0);
#else
    asm volatile("s_wait_asynccnt 0" ::: "memory");
#endif
  }
#else
  for (int j = tid; j < LBL_; j += 256) s_neg[j] = neg_scores[rowbase + j];
#endif

  // rank_weights: rw[0]=1, rw[r]=rw[r-1] + 1 + 1/r (sequential f32 cumsum).
  if (tid == 0) {
    float rw = 0.0f;
    for (int r = 0; r < 256; ++r) {
      rw += (r == 0) ? 1.0f : (1.0f + 1.0f / (float)r);
      s_rw[r] = rw;
    }
  }
  __syncthreads();

  const int   nn   = n_neg[i];
  const float fnn  = (float)nn;
  const int   kmax = (nn > 0) ? (nn - 1) : 0;

  float acc = 0.0f;  // uniform across the wave
  for (int j = wv; j < LBL_; j += 8) {
    if (s_t[j] != 1) continue;        // L is zero unless target==1
    const float  x  = s_x[j];
    const size_t ub = (rowbase + (size_t)j) * (size_t)T_;

    int nt = T_;                      // default: no success in T trials
    for (int b0 = 0; b0 < T_; b0 += 32) {
      const int t    = b0 + lane;
      bool      succ = false;
      if (t < T_) {
        const float uu = u[ub + (size_t)t];
        int k = (int)(uu * fnn);      // trunc-toward-zero == astype(int32)
        k = (k > kmax) ? kmax : k;
        if (k < 0) k = 0;
        succ = (s_neg[k] - x) >= 0.0f;
      }
      const unsigned long long m = __ballot(succ);
      if (m != 0ull) {
        nt = b0 + __ffsll(m);         // first success index + 1
        break;
      }
      if (b0 + 32 < T_)               // speculative prefetch of next chunk
        __builtin_prefetch((const void*)(u + ub + (size_t)(b0 + 32)), 0, 0);
    }
    acc += s_rw[T_ / nt];             // r in [1,255]
  }

  if (lane == 0) s_part[wv] = acc;
  __syncthreads();
  if (tid == 0) {
    float s = 0.0f;
    for (int w = 0; w < 8; ++w) s += s_part[w];
    row_loss[i] = s * margin_term[i];  // margin_term factors out of sum_j
  }
}

// ---------------------------------------------------------------------------
// Kernel 3: deterministic reduction of 128 row losses -> scalar.
// ---------------------------------------------------------------------------
__global__ __launch_bounds__(128) void warp_reduce(
    const float* __restrict__ row_loss, float* __restrict__ out) {
  __shared__ float s[B_];
  const int tid = threadIdx.x;
  s[tid] = row_loss[tid];
  __syncthreads();
  for (int off = B_ / 2; off > 0; off >>= 1) {
    if (tid < off) s[tid] += s[tid + off];
    __syncthreads();
  }
  if (tid == 0) out[0] = s[0];
}

extern "C" void kernel_launch(void* const* d_in, const int* in_sizes, int n_in,
                              void* d_out, int out_size, void* d_ws, size_t ws_size,
                              hipStream_t stream) {
  (void)in_sizes; (void)n_in; (void)out_size; (void)ws_size;
  const float* inp = (const float*)d_in[0];  // [B, LBL] f32
  const float* u   = (const float*)d_in[1];  // [B, LBL, T] f32
  const int*   tg  = (const int*)d_in[2];    // [B, LBL] i32
  float*       out = (float*)d_out;          // scalar

  char*  ws         = (char*)d_ws;
  float* neg_scores = (float*)ws;                                  // B*LBL f32
  float* margin     = (float*)(ws + (size_t)B_ * LBL_ * 4);        // B f32
  int*   nneg       = (int*)(ws + (size_t)B_ * LBL_ * 4 + B_ * 4); // B i32
  float* rloss      = (float*)(ws + (size_t)B_ * LBL_ * 4 + 2 * (size_t)B_ * 4);

  warp_prep<<<B_, 256, 0, stream>>>(inp, tg, neg_scores, nneg);
  warp_margin<<<B_ / 16, 32, 0, stream>>>(inp, tg, margin);
  warp_main<<<B_, 256, 0, stream>>>(tg, inp, u, neg_scores, margin, nneg, rloss);
  warp_reduce<<<1, B_, 0, stream>>>(rloss, out);
}